// DCNNLoss_81209241633392
// MI455X (gfx1250) — compile-verified
//
#include <hip/hip_runtime.h>
#include <hip/hip_bf16.h>

// ---------------------------------------------------------------------------
// DCNN loss (CE + pairwise hinge) for MI455X / gfx1250, wave32.
//
// B = 131072 rows, C = 256 cols, fp32. Memory-bound: 128 MB @ 23.3 TB/s
// => ~5.6 us floor, single pass over the input.
//
// Design:
//  * 1 wave (32 threads) per 16-row tile (=> 8 consecutive row-pairs).
//  * Tile staged global->LDS with CDNA5 async copies
//    (GLOBAL_LOAD_ASYNC_TO_LDS_B128, ASYNCcnt) -- no VGPR round-trip,
//    fenced with s_wait_asynccnt 0.
//  * LDS rows padded (+4 floats, stride 260) so CE reads (b128) and WMMA
//    fragment reads (b64) are bank-conflict-free.
//  * Gram matrix G = X X^T of the tile via 64 chained
//    V_WMMA_F32_16X16X4_F32 ops (A fragment == B fragment by symmetry).
//    d2 for pair p reconstructed analytically from G and row sums:
//      d2 = G00/m0^2 + G11/m1^2 - 2 G01/(m0 m1)
//         + 2 eps (s0/m0 - s1/m1) + C eps^2,   m = max(sqrt(Gkk), NORM_EPS)
//  * CE per row with wave32 shuffle reductions (max, sum exp, row sum,
//    label-element pick).
//  * cumsum(eq) > 0 == (pair_idx >= first index with equal labels):
//    atomicMin of the first equal-pair index (deterministic).
//  * Final single-block kernel reduces ce/hinge in a fixed order.
// ---------------------------------------------------------------------------

#define COLS      256
#define CLS       260          // padded LDS row stride in floats
#define LAMDA_F   0.05f
#define TAU_F     0.44f
#define MARGIN_F  0.05f
#define PD_EPS_F  1e-06f
#define NORM_EPS_F 1e-12f

typedef __attribute__((ext_vector_type(2))) float v2f;
typedef __attribute__((ext_vector_type(8))) float v8f;

__global__ __launch_bounds__(1) void k_init(int* min_eq) {
    *min_eq = 0x7FFFFFFF;
}

__global__ __launch_bounds__(32) void k_main(const float* __restrict__ inp,
                                             const int*   __restrict__ labels,
                                             float* __restrict__ ce_pair,
                                             float* __restrict__ d2_pair,
                                             int*   __restrict__ min_eq) {
    __shared__ float lds[16 * CLS];
    __shared__ float gmat[16][17];
    __shared__ float ce_s[16];
    __shared__ float sum_s[16];

    const int lane = threadIdx.x;          // 0..31 (wave32)
    const int t    = blockIdx.x;           // tile index; rows 16t..16t+15
    const float* base = inp + (size_t)t * 16 * COLS;

    // ---- Stage 16 rows (16 KB) into padded LDS via async copies --------
    // Generic pointers into LDS carry the wave-relative LDS byte offset in
    // their low 32 bits.
    const unsigned int      lbase = (unsigned int)(unsigned long long)(uintptr_t)&lds[0];
    const unsigned long long gbase = (unsigned long long)(uintptr_t)base;
    #pragma unroll
    for (int i = 0; i < 32; ++i) {
        const int row  = i >> 1;
        const int colb = ((i & 1) << 7) + lane * 4;   // half-row per instr
        const unsigned int      loff = lbase + (unsigned int)((row * CLS + colb) * 4);
        const unsigned long long ga  = gbase + (unsigned long long)((row * COLS + colb) * 4);
        asm volatile("global_load_async_to_lds_b128 %0, %1, off"
                     :: "v"(loff), "v"(ga) : "memory");
    }
    asm volatile("s_wait_asynccnt 0x0" ::: "memory");
    __syncthreads();

    // ---- CE pass: per-row logsumexp, label pick, row sum ---------------
    for (int r = 0; r < 16; ++r) {
        const float4 p0 = *(const float4*)&lds[r * CLS + lane * 8];
        const float4 p1 = *(const float4*)&lds[r * CLS + lane * 8 + 4];
        float x[8] = {p0.x, p0.y, p0.z, p0.w, p1.x, p1.y, p1.z, p1.w};

        float mx = x[0];
        #pragma unroll
        for (int j = 1; j < 8; ++j) mx = fmaxf(mx, x[j]);
        #pragma unroll
        for (int o = 16; o > 0; o >>= 1) mx = fmaxf(mx, __shfl_xor(mx, o, 32));

        const int lab = labels[t * 16 + r];     // int32 (jax default int)
        float se = 0.0f, sm = 0.0f, lv = 0.0f;
        #pragma unroll
        for (int j = 0; j < 8; ++j) { se += __expf(x[j] - mx); sm += x[j]; }
        if ((lab >> 3) == lane) lv = x[lab & 7];

        #pragma unroll
        for (int o = 16; o > 0; o >>= 1) {
            se += __shfl_xor(se, o, 32);
            sm += __shfl_xor(sm, o, 32);
            lv += __shfl_xor(lv, o, 32);
        }
        if (lane == 0) {
            ce_s[r]  = mx + __logf(se) - lv;    // logsumexp - x[label]
            sum_s[r] = sm;
        }
    }

    // ---- Gram matrix G = X X^T via V_WMMA_F32_16X16X4_F32 --------------
    // A fragment (16x4 f32, 2 VGPRs): lanes 0-15 hold K={0,1} of row M=lane,
    // lanes 16-31 hold K={2,3} of row M=lane-16. B fragment of X^T chunk has
    // the identical per-lane data (symmetry), so one ds_read_b64 feeds both.
    v8f acc = {};
    const int m = lane & 15;
    const int h = lane >> 4;
    for (int k = 0; k < COLS; k += 4) {
        v2f f = *(const v2f*)&lds[m * CLS + k + 2 * h];
        acc = __builtin_amdgcn_wmma_f32_16x16x4_f32(
                  false, f, false, f, (short)0, acc, false, false);
    }
    // C/D layout: VGPR v, lanes 0-15 -> M=v, lanes 16-31 -> M=v+8; N=lane&15
    #pragma unroll
    for (int v = 0; v < 8; ++v) gmat[v + 8 * h][m] = acc[v];
    __syncthreads();

    // ---- Per-pair d2 + eq + outputs (lanes 0..7 own pairs 0..7) --------
    if (lane < 8) {
        const int p   = lane;
        const int r0  = 2 * p, r1 = 2 * p + 1;
        const float g00 = gmat[r0][r0];
        const float g11 = gmat[r1][r1];
        const float g01 = gmat[r0][r1];
        const float m0 = fmaxf(sqrtf(g00), NORM_EPS_F);
        const float m1 = fmaxf(sqrtf(g11), NORM_EPS_F);
        const float d2 = g00 / (m0 * m0) + g11 / (m1 * m1)
                       - 2.0f * g01 / (m0 * m1)
                       + 2.0f * PD_EPS_F * (sum_s[r0] / m0 - sum_s[r1] / m1)
                       + (float)COLS * PD_EPS_F * PD_EPS_F;
        const int pg = t * 8 + p;
        d2_pair[pg] = d2;
        ce_pair[pg] = ce_s[r0] + ce_s[r1];
        const int l0 = labels[t * 16 + r0];
        const int l1 = labels[t * 16 + r1];
        if (l0 == l1) atomicMin(min_eq, pg);
    }
}

__global__ __launch_bounds__(256) void k_final(const float* __restrict__ ce_pair,
                                               const float* __restrict__ d2_pair,
                                               const int*   __restrict__ min_eq,
                                               float* __restrict__ out,
                                               int nPairs, int nRows) {
    __shared__ float sh_ce[256];
    __shared__ float sh_h[256];
    const int tid = threadIdx.x;
    const int me  = *min_eq;

    float ac = 0.0f, ah = 0.0f;
    for (int i = tid; i < nPairs; i += 256) {
        ac += ce_pair[i];
        const float l = (i >= me) ? 1.0f : -1.0f;
        const float hv = MARGIN_F - l * (TAU_F - d2_pair[i]);
        ah += fmaxf(hv, 0.0f);
    }
    sh_ce[tid] = ac;
    sh_h[tid]  = ah;
    __syncthreads();
    #pragma unroll
    for (int s = 128; s > 0; s >>= 1) {
        if (tid < s) { sh_ce[tid] += sh_ce[tid + s]; sh_h[tid] += sh_h[tid + s]; }
        __syncthreads();
    }
    if (tid == 0)
        out[0] = sh_ce[0] / (float)nRows + LAMDA_F * sh_h[0] * 0.5f;
}

extern "C" void kernel_launch(void* const* d_in, const int* in_sizes, int n_in,
                              void* d_out, int out_size, void* d_ws, size_t ws_size,
                              hipStream_t stream) {
    const float* inp    = (const float*)d_in[0];
    const int*   labels = (const int*)d_in[1];   // randint downcasts to int32 by default
    float*       out    = (float*)d_out;

    const int nRows  = in_sizes[0] / COLS;       // 131072
    const int nPairs = nRows / 2;                // 65536
    const int nTiles = nRows / 16;               // 8192

    int*   min_eq  = (int*)d_ws;
    float* ce_pair = (float*)((char*)d_ws + 256);
    float* d2_pair = ce_pair + nPairs;           // ws use: 256B + 512KB

    k_init <<<1, 1, 0, stream>>>(min_eq);
    k_main <<<nTiles, 32, 0, stream>>>(inp, labels, ce_pair, d2_pair, min_eq);
    k_final<<<1, 256, 0, stream>>>(ce_pair, d2_pair, min_eq, out, nPairs, nRows);
}